// LightGCN_55929064129244
// MI455X (gfx1250) — compile-verified
//
#include <hip/hip_runtime.h>
#include <math.h>

// Problem sizes (fixed by reference)
#define U_N 100000
#define I_N 50000
#define E_N 2000000
#define D_N 128
#define B_N 4096
#define CHUNK 1024   // scan chunk (256 threads x 4)

typedef float v2f __attribute__((ext_vector_type(2)));
typedef float v8f __attribute__((ext_vector_type(8)));

__global__ void k_zero(float* __restrict__ p, long n) {
  long t = (long)blockIdx.x * blockDim.x + threadIdx.x;
  if (t < n) p[t] = 0.0f;
}

// -------- CSR build --------

__global__ void k_deg(const int* __restrict__ u, const int* __restrict__ i,
                      int* __restrict__ du, int* __restrict__ di) {
  int e = blockIdx.x * blockDim.x + threadIdx.x;
  if (e < E_N) {
    atomicAdd(&du[u[e]], 1);
    atomicAdd(&di[i[e]], 1);
  }
}

// Pass A: per-chunk sums
__global__ void k_chunksum(const int* __restrict__ deg, int* __restrict__ bsum, int n) {
  __shared__ int red[256];
  int c0 = blockIdx.x * CHUNK;
  int tid = threadIdx.x;
  int s = 0;
#pragma unroll
  for (int j = 0; j < 4; ++j) {
    int idx = c0 + tid * 4 + j;
    if (idx < n) s += deg[idx];
  }
  red[tid] = s;
  __syncthreads();
  for (int st = 128; st > 0; st >>= 1) {
    if (tid < st) red[tid] += red[tid + st];
    __syncthreads();
  }
  if (tid == 0) bsum[blockIdx.x] = red[0];
}

// Pass B: serial exclusive scan of <=98 chunk sums
__global__ void k_scanblocks(int* __restrict__ bsum, int nc, int* __restrict__ off_total) {
  if (blockIdx.x == 0 && threadIdx.x == 0) {
    int acc = 0;
    for (int c = 0; c < nc; ++c) { int t = bsum[c]; bsum[c] = acc; acc += t; }
    *off_total = acc;   // off[n] = E
  }
}

// Pass C: final exclusive offsets
__global__ void k_scanfinal(const int* __restrict__ deg, const int* __restrict__ bsum,
                            int* __restrict__ off, int n) {
  __shared__ int tsum[256];
  __shared__ int tpre[256];
  int c0 = blockIdx.x * CHUNK;
  int tid = threadIdx.x;
  int v[4];
  int s = 0;
#pragma unroll
  for (int j = 0; j < 4; ++j) {
    int idx = c0 + tid * 4 + j;
    v[j] = (idx < n) ? deg[idx] : 0;
    s += v[j];
  }
  tsum[tid] = s;
  __syncthreads();
  if (tid == 0) {
    int acc = 0;
    for (int k = 0; k < 256; ++k) { tpre[k] = acc; acc += tsum[k]; }
  }
  __syncthreads();
  int p = bsum[blockIdx.x] + tpre[tid];
#pragma unroll
  for (int j = 0; j < 4; ++j) {
    int idx = c0 + tid * 4 + j;
    if (idx < n) off[idx] = p;
    p += v[j];
  }
}

// Scatter edges into both adjacency lists; bake edge weight in.
__global__ void k_fill(const int* __restrict__ u_idx, const int* __restrict__ i_idx,
                       const int* __restrict__ du, const int* __restrict__ di,
                       const int* __restrict__ off_u, const int* __restrict__ off_i,
                       int* __restrict__ cur_u, int* __restrict__ cur_i,
                       int* __restrict__ nbr_u, float* __restrict__ w_u,
                       int* __restrict__ nbr_i, float* __restrict__ w_i) {
  int e = blockIdx.x * blockDim.x + threadIdx.x;
  if (e >= E_N) return;
  int u = u_idx[e], i = i_idx[e];
  float w = rsqrtf((float)du[u] * (float)di[i]);  // counts < 2^24: exact
  int pu = off_u[u] + atomicAdd(&cur_u[u], 1);
  nbr_u[pu] = i; w_u[pu] = w;
  int pi = off_i[i] + atomicAdd(&cur_i[i], 1);
  nbr_i[pi] = u; w_i[pi] = w;
}

// -------- Propagation: pull-mode, one wave per node, no fp atomics --------
// hdst[node] = hself[node] + sum_k w[k] * hother[nbr[k]]
__global__ void k_node(const float* __restrict__ hself, const float* __restrict__ hother,
                       float* __restrict__ hdst, const int* __restrict__ off,
                       const int* __restrict__ nbr, const float* __restrict__ w,
                       int n_nodes) {
  int wid = (int)(((long)blockIdx.x * blockDim.x + threadIdx.x) >> 5);
  if (wid >= n_nodes) return;
  int lane = threadIdx.x & 31;
  int c = lane << 2;                         // 32 lanes x float4 = 128-float row
  int s = off[wid], e = off[wid + 1];
  float4 acc = *(const float4*)(hself + (long)wid * D_N + c);
  for (int k = s; k < e; ++k) {
    int nb = nbr[k];                         // wave-uniform
    float ww = w[k];
    float4 hv = *(const float4*)(hother + (long)nb * D_N + c);
    acc.x += ww * hv.x;
    acc.y += ww * hv.y;
    acc.z += ww * hv.z;
    acc.w += ww * hv.w;
  }
  *(float4*)(hdst + (long)wid * D_N + c) = acc;
}

// dst[b,:] (acc?+=:=) coeff * src[idx[b],:]
__global__ void k_gather(const float* __restrict__ src, const int* __restrict__ idx,
                         float* __restrict__ dst, float coeff, int accumulate) {
  int t = blockIdx.x * blockDim.x + threadIdx.x;
  if (t >= B_N * D_N) return;
  int b = t >> 7;
  int d = t & (D_N - 1);
  float v = coeff * src[(long)idx[b] * D_N + d];
  dst[t] = accumulate ? (dst[t] + v) : v;
}

// -------- WMMA scoring (verified lowering: v_wmma_f32_16x16x4_f32 chain) --------
__global__ void __launch_bounds__(32)
k_score(const float* __restrict__ ue, const float* __restrict__ ie,
        const float* __restrict__ labels, float* __restrict__ out,
        float* __restrict__ lossacc) {
  __shared__ float lds[256];
  int lane = threadIdx.x;
  int tile = blockIdx.x;
  int r = lane & 15;
  int half = lane >> 4;
  const float* ua = ue + (long)(tile * 16 + r) * D_N;
  const float* ib = ie + (long)(tile * 16 + r) * D_N;
  v8f c = {};
  for (int k0 = 0; k0 < D_N; k0 += 4) {
    int k = k0 + (half << 1);
    v2f a, b;
    a[0] = ua[k];
    a[1] = ua[k + 1];
    b[0] = ib[k];
    b[1] = ib[k + 1];
    c = __builtin_amdgcn_wmma_f32_16x16x4_f32(false, a, false, b, (short)0, c,
                                              false, false);
  }
#pragma unroll
  for (int q = 0; q < 8; ++q) lds[q * 32 + lane] = c[q];
  __syncthreads();
  if (lane < 16) {
    float s = (lane < 8) ? lds[lane * 32 + lane]
                         : lds[(lane - 8) * 32 + 16 + lane];
    int bi = tile * 16 + lane;
    float z = 1.0f / (1.0f + expf(-s));
    out[1 + bi] = z;
    float lab = labels[bi];
    float term = fmaxf(z, 0.0f) - z * lab + log1pf(expf(-fabsf(z)));
    atomicAdd(lossacc, term);
  }
}

__global__ void k_finalize(const float* __restrict__ lossacc, float* __restrict__ out) {
  out[0] = lossacc[0] * (1.0f / (float)B_N);
}

extern "C" void kernel_launch(void* const* d_in, const int* in_sizes, int n_in,
                              void* d_out, int out_size, void* d_ws, size_t ws_size,
                              hipStream_t stream) {
  (void)in_sizes; (void)n_in; (void)out_size; (void)ws_size;
  const float* user_feat    = (const float*)d_in[0];
  const float* item_feat    = (const float*)d_in[1];
  const int*   u_idx        = (const int*)d_in[2];
  const int*   i_idx        = (const int*)d_in[3];
  const int*   user_indices = (const int*)d_in[4];
  const int*   item_indices = (const int*)d_in[5];
  const float* labels       = (const float*)d_in[6];
  float* out = (float*)d_out;

  // Workspace carve-up (4-byte words), ~48M words ~ 192 MB.
  // 16B-aligned arrays first.
  char* base = (char*)d_ws;
  long off = 0;
  auto carve = [&](long words) { void* p = base + off * 4; off += words; return p; };
  float* hu_a = (float*)carve((long)U_N * D_N);
  float* hu_b = (float*)carve((long)U_N * D_N);
  float* hi_a = (float*)carve((long)I_N * D_N);
  float* hi_b = (float*)carve((long)I_N * D_N);
  int*   nbr_u = (int*)carve(E_N);
  float* w_u   = (float*)carve(E_N);
  int*   nbr_i = (int*)carve(E_N);
  float* w_i   = (float*)carve(E_N);
  float* ueg   = (float*)carve((long)B_N * D_N);
  float* ieg   = (float*)carve((long)B_N * D_N);
  int*   off_u = (int*)carve(U_N + 1);
  int*   off_i = (int*)carve(I_N + 1);
  // contiguous zero region: deg_u, deg_i, cur_u, cur_i, bsum, lossacc
  long zstart = off;
  int*   deg_u = (int*)carve(U_N);
  int*   deg_i = (int*)carve(I_N);
  int*   cur_u = (int*)carve(U_N);
  int*   cur_i = (int*)carve(I_N);
  int*   bsum  = (int*)carve(128);
  float* lossacc = (float*)carve(1);
  long znum = off - zstart;

  const int TB = 256;
  auto nb = [](long n, int tb) { return (unsigned)((n + tb - 1) / tb); };
  const int ncU = (U_N + CHUNK - 1) / CHUNK;  // 98
  const int ncI = (I_N + CHUNK - 1) / CHUNK;  // 49

  // zero deg/cur/bsum/lossacc
  k_zero<<<nb(znum, TB), TB, 0, stream>>>((float*)deg_u, znum);

  // degrees + exclusive scans -> CSR offsets
  k_deg<<<nb(E_N, TB), TB, 0, stream>>>(u_idx, i_idx, deg_u, deg_i);
  k_chunksum<<<ncU, TB, 0, stream>>>(deg_u, bsum, U_N);
  k_scanblocks<<<1, 1, 0, stream>>>(bsum, ncU, off_u + U_N);
  k_scanfinal<<<ncU, TB, 0, stream>>>(deg_u, bsum, off_u, U_N);
  k_chunksum<<<ncI, TB, 0, stream>>>(deg_i, bsum, I_N);
  k_scanblocks<<<1, 1, 0, stream>>>(bsum, ncI, off_i + I_N);
  k_scanfinal<<<ncI, TB, 0, stream>>>(deg_i, bsum, off_i, I_N);

  // fill adjacency (weights baked in)
  k_fill<<<nb(E_N, TB), TB, 0, stream>>>(u_idx, i_idx, deg_u, deg_i, off_u, off_i,
                                         cur_u, cur_i, nbr_u, w_u, nbr_i, w_i);

  // h0 = features
  hipMemcpyAsync(hu_a, user_feat, (size_t)U_N * D_N * sizeof(float),
                 hipMemcpyDeviceToDevice, stream);
  hipMemcpyAsync(hi_a, item_feat, (size_t)I_N * D_N * sizeof(float),
                 hipMemcpyDeviceToDevice, stream);

  // ue_g/ie_g layer-0 contribution (coeff 1)
  k_gather<<<nb((long)B_N * D_N, TB), TB, 0, stream>>>(user_feat, user_indices, ueg, 1.0f, 0);
  k_gather<<<nb((long)B_N * D_N, TB), TB, 0, stream>>>(item_feat, item_indices, ieg, 1.0f, 0);

  // 3 propagation layers: pull-mode, double-buffered, zero fp atomics
  float* hu_s = hu_a; float* hu_d = hu_b;
  float* hi_s = hi_a; float* hi_d = hi_b;
  for (int l = 0; l < 3; ++l) {
    k_node<<<nb((long)U_N * 32, TB), TB, 0, stream>>>(hu_s, hi_s, hu_d, off_u, nbr_u, w_u, U_N);
    k_node<<<nb((long)I_N * 32, TB), TB, 0, stream>>>(hi_s, hu_s, hi_d, off_i, nbr_i, w_i, I_N);
    float* t;
    t = hu_s; hu_s = hu_d; hu_d = t;
    t = hi_s; hi_s = hi_d; hi_d = t;
    float coeff = 1.0f / (float)(l + 2);
    k_gather<<<nb((long)B_N * D_N, TB), TB, 0, stream>>>(hu_s, user_indices, ueg, coeff, 1);
    k_gather<<<nb((long)B_N * D_N, TB), TB, 0, stream>>>(hi_s, item_indices, ieg, coeff, 1);
  }

  // WMMA scoring + loss
  k_score<<<B_N / 16, 32, 0, stream>>>(ueg, ieg, labels, out, lossacc);
  k_finalize<<<1, 1, 0, stream>>>(lossacc, out);
}